// LearnableWeightedRGCN2_42245298323706
// MI455X (gfx1250) — compile-verified
//
#include <hip/hip_runtime.h>
#include <math.h>

#define NN 30000
#define DD 128
#define RR 8
#define BB 4
#define EE 480000
#define LN_EPS 1e-5f

typedef __attribute__((ext_vector_type(2))) float v2f;
typedef __attribute__((ext_vector_type(8))) float v8f;

// ---------------------------------------------------------------------------
// w_all[r] = sum_b coeffs[r,b] * bases[b]   ([R,D,D])
// ---------------------------------------------------------------------------
__global__ __launch_bounds__(256) void compose_kernel(
    const float* __restrict__ coeffs,  // [R,B]
    const float* __restrict__ bases,   // [B,D,D]
    float* __restrict__ w_all)         // [R,D,D]
{
    int idx = blockIdx.x * 256 + threadIdx.x;      // < R*D*D = 131072
    int r   = idx >> 14;                            // D*D = 16384
    int off = idx & 16383;
    float acc = 0.0f;
#pragma unroll
    for (int b = 0; b < BB; ++b)
        acc += coeffs[r * BB + b] * bases[b * (DD * DD) + off];
    w_all[idx] = acc;
}

// ---------------------------------------------------------------------------
// C = A @ W^T (+bias):  C[m,j] = sum_k A[m,k] * W[j,k]
// fp32 WMMA 16x16x4. Block = 256 thr = 8 waves, tile 64 rows x 128 cols.
// Wave w: row-tile (w&3), col range 64*(w>>2). blockIdx.z selects W/C slice.
// ---------------------------------------------------------------------------
#define TILE_M 64
#define LDSW 132   // 128 + 4 pad

__global__ __launch_bounds__(256) void gemm_xwt_kernel(
    const float* __restrict__ A,     // [M,128]
    const float* __restrict__ Wt,    // [128,128] (+ z*strideW)
    const float* __restrict__ bias,  // [128] or unused
    float* __restrict__ C,           // [M,128]   (+ z*strideC)
    int M, int hasBias, int strideW, long strideC)
{
    __shared__ float As[TILE_M * LDSW];

    const float* W = Wt + (size_t)blockIdx.z * strideW;
    float* Cz      = C  + (size_t)blockIdx.z * strideC;

    const int tid    = threadIdx.x;
    const int wv     = tid >> 5;
    const int lane   = tid & 31;
    const int rowBlk = blockIdx.x * TILE_M;

    // stage A block (64x128) into LDS, coalesced float4
    for (int i = tid; i < TILE_M * 32; i += 256) {
        int r = i >> 5;
        int c4 = (i & 31) * 4;
        int grow = rowBlk + r;
        float4 v = make_float4(0.f, 0.f, 0.f, 0.f);
        if (grow < M) v = *(const float4*)(A + (size_t)grow * DD + c4);
        *(float4*)(&As[r * LDSW + c4]) = v;
    }
    __syncthreads();

    const int rt    = wv & 3;           // row tile
    const int c0    = (wv >> 2) * 64;   // col base
    const int lhalf = lane & 15;        // = M (A) / N (B)
    const int khalf = (lane >> 4) * 2;  // 0 or 2

    v8f acc0 = {}, acc1 = {}, acc2 = {}, acc3 = {};
    for (int kb = 0; kb < DD; kb += 4) {
        v2f a = *(const v2f*)(&As[(rt * 16 + lhalf) * LDSW + kb + khalf]);
        v2f b0 = *(const v2f*)(W + (size_t)(c0 +  0 + lhalf) * DD + kb + khalf);
        v2f b1 = *(const v2f*)(W + (size_t)(c0 + 16 + lhalf) * DD + kb + khalf);
        v2f b2 = *(const v2f*)(W + (size_t)(c0 + 32 + lhalf) * DD + kb + khalf);
        v2f b3 = *(const v2f*)(W + (size_t)(c0 + 48 + lhalf) * DD + kb + khalf);
        acc0 = __builtin_amdgcn_wmma_f32_16x16x4_f32(false, a, false, b0, (short)0, acc0, false, false);
        acc1 = __builtin_amdgcn_wmma_f32_16x16x4_f32(false, a, false, b1, (short)0, acc1, false, false);
        acc2 = __builtin_amdgcn_wmma_f32_16x16x4_f32(false, a, false, b2, (short)0, acc2, false, false);
        acc3 = __builtin_amdgcn_wmma_f32_16x16x4_f32(false, a, false, b3, (short)0, acc3, false, false);
    }

    v8f accs[4] = {acc0, acc1, acc2, acc3};
#pragma unroll
    for (int t = 0; t < 4; ++t) {
        int col = c0 + t * 16 + lhalf;
#pragma unroll
        for (int i = 0; i < 8; ++i) {
            int row = rowBlk + rt * 16 + ((lane < 16) ? i : i + 8);
            if (row < M) {
                float v = accs[t][i];
                if (hasBias) v += bias[col];
                Cz[(size_t)row * DD + col] = v;
            }
        }
    }
}

// ---------------------------------------------------------------------------
// Row-wise LayerNorm, one wave per row (lane owns 4 contiguous floats)
// ---------------------------------------------------------------------------
__global__ __launch_bounds__(256) void layernorm_kernel(
    const float* __restrict__ in, const float* __restrict__ g,
    const float* __restrict__ b, float* __restrict__ out, int M)
{
    int wv = threadIdx.x >> 5, lane = threadIdx.x & 31;
    int row = blockIdx.x * 8 + wv;
    if (row >= M) return;
    const float* p = in + (size_t)row * DD + lane * 4;
    float x0 = p[0], x1 = p[1], x2 = p[2], x3 = p[3];
    float s = x0 + x1 + x2 + x3;
    for (int o = 16; o; o >>= 1) s += __shfl_xor(s, o, 32);
    float mu = s * (1.0f / DD);
    float d0 = x0 - mu, d1 = x1 - mu, d2 = x2 - mu, d3 = x3 - mu;
    float q = d0 * d0 + d1 * d1 + d2 * d2 + d3 * d3;
    for (int o = 16; o; o >>= 1) q += __shfl_xor(q, o, 32);
    float rs = rsqrtf(q * (1.0f / DD) + LN_EPS);
    const float* gp = g + lane * 4;
    const float* bp = b + lane * 4;
    float* op = out + (size_t)row * DD + lane * 4;
    op[0] = d0 * rs * gp[0] + bp[0];
    op[1] = d1 * rs * gp[1] + bp[1];
    op[2] = d2 * rs * gp[2] + bp[2];
    op[3] = d3 * rs * gp[3] + bp[3];
}

// ---------------------------------------------------------------------------
// SpMM: hout[row] += val * hin[col].  One wave per edge, 4 edges per wave.
// hout must be pre-zeroed. f32 atomics -> global_atomic_add_f32 (no-return).
// ---------------------------------------------------------------------------
__global__ __launch_bounds__(256) void spmm_kernel(
    const float* __restrict__ hin, float* __restrict__ hout,
    const int* __restrict__ rows, const int* __restrict__ cols,
    const float* __restrict__ vals, int nEdges)
{
    int lane = threadIdx.x & 31;
    int wv   = threadIdx.x >> 5;
    int base = (blockIdx.x * 8 + wv) * 4;
#pragma unroll
    for (int j = 0; j < 4; ++j) {
        int e = base + j;
        if (e >= nEdges) break;
        int   row = rows[e];
        int   col = cols[e];
        float v   = vals[e];
        const float* ip = hin + (size_t)col * DD + lane * 4;
        float i0 = ip[0], i1 = ip[1], i2 = ip[2], i3 = ip[3];
        float* op = hout + (size_t)row * DD + lane * 4;
        unsafeAtomicAdd(op + 0, v * i0);
        unsafeAtomicAdd(op + 1, v * i1);
        unsafeAtomicAdd(op + 2, v * i2);
        unsafeAtomicAdd(op + 3, v * i3);
    }
}

// ---------------------------------------------------------------------------
// Fused attention: tanh-score, mask, softmax, weighted sum, bias, GELU, LN.
// One wave per node; lane owns 4 contiguous channels of all 8 relations.
// ---------------------------------------------------------------------------
__global__ __launch_bounds__(256) void attn_out_kernel(
    const float* __restrict__ stacked,   // [R][N][128]
    const float* __restrict__ attn_vec,  // [128]
    const float* __restrict__ bias,      // [128]
    const float* __restrict__ fg, const float* __restrict__ fb,
    float* __restrict__ outFinal,        // [N,128]
    float* __restrict__ outAttn,         // [N,8]
    int M)
{
    int wv = threadIdx.x >> 5, lane = threadIdx.x & 31;
    int n = blockIdx.x * 8 + wv;
    if (n >= M) return;

    const float* avp = attn_vec + lane * 4;
    float a0 = avp[0], a1 = avp[1], a2 = avp[2], a3 = avp[3];

    float s0[RR], s1[RR], s2[RR], s3[RR], sc[RR];
    int valid[RR];
#pragma unroll
    for (int r = 0; r < RR; ++r) {
        const float* sp = stacked + ((size_t)r * M + n) * DD + lane * 4;
        s0[r] = sp[0]; s1[r] = sp[1]; s2[r] = sp[2]; s3[r] = sp[3];
        float p = tanhf(s0[r]) * a0 + tanhf(s1[r]) * a1 +
                  tanhf(s2[r]) * a2 + tanhf(s3[r]) * a3;
        int f = (s0[r] != 0.f) | (s1[r] != 0.f) | (s2[r] != 0.f) | (s3[r] != 0.f);
        for (int o = 16; o; o >>= 1) {
            p += __shfl_xor(p, o, 32);
            f |= __shfl_xor(f, o, 32);
        }
        sc[r] = p; valid[r] = f;
    }

    float m = -INFINITY;
#pragma unroll
    for (int r = 0; r < RR; ++r) if (valid[r]) m = fmaxf(m, sc[r]);
    float ev[RR], sum = 0.f;
#pragma unroll
    for (int r = 0; r < RR; ++r) {
        ev[r] = valid[r] ? __expf(sc[r] - m) : 0.f;
        sum += ev[r];
    }
    float inv = (sum > 0.f) ? (1.0f / sum) : 0.f;  // all-masked -> zeros (nan_to_num)
    float aw[RR];
#pragma unroll
    for (int r = 0; r < RR; ++r) aw[r] = ev[r] * inv;

    const float* bp = bias + lane * 4;
    float o0 = bp[0], o1 = bp[1], o2 = bp[2], o3 = bp[3];
#pragma unroll
    for (int r = 0; r < RR; ++r) {
        o0 += aw[r] * s0[r]; o1 += aw[r] * s1[r];
        o2 += aw[r] * s2[r]; o3 += aw[r] * s3[r];
    }
    // exact GELU: 0.5*x*(1+erf(x/sqrt(2)))
    o0 = 0.5f * o0 * (1.0f + erff(o0 * 0.70710678118654752f));
    o1 = 0.5f * o1 * (1.0f + erff(o1 * 0.70710678118654752f));
    o2 = 0.5f * o2 * (1.0f + erff(o2 * 0.70710678118654752f));
    o3 = 0.5f * o3 * (1.0f + erff(o3 * 0.70710678118654752f));

    // final LayerNorm
    float s = o0 + o1 + o2 + o3;
    for (int o = 16; o; o >>= 1) s += __shfl_xor(s, o, 32);
    float mu = s * (1.0f / DD);
    float d0 = o0 - mu, d1 = o1 - mu, d2 = o2 - mu, d3 = o3 - mu;
    float q = d0 * d0 + d1 * d1 + d2 * d2 + d3 * d3;
    for (int o = 16; o; o >>= 1) q += __shfl_xor(q, o, 32);
    float rs = rsqrtf(q * (1.0f / DD) + LN_EPS);
    const float* gp = fg + lane * 4;
    const float* fbp = fb + lane * 4;
    float* op = outFinal + (size_t)n * DD + lane * 4;
    op[0] = d0 * rs * gp[0] + fbp[0];
    op[1] = d1 * rs * gp[1] + fbp[1];
    op[2] = d2 * rs * gp[2] + fbp[2];
    op[3] = d3 * rs * gp[3] + fbp[3];

    if (lane == 0) {
        float* ap = outAttn + (size_t)n * RR;
#pragma unroll
        for (int r = 0; r < RR; ++r) ap[r] = aw[r];
    }
}

// ---------------------------------------------------------------------------
extern "C" void kernel_launch(void* const* d_in, const int* in_sizes, int n_in,
                              void* d_out, int out_size, void* d_ws, size_t ws_size,
                              hipStream_t stream) {
    const float* node_emb = (const float*)d_in[0];   // [N,D]
    const float* pre_W    = (const float*)d_in[1];   // [D,D]
    const float* pre_b    = (const float*)d_in[2];   // [D]
    const float* ln1_g    = (const float*)d_in[3];
    const float* ln1_b    = (const float*)d_in[4];
    const float* bases    = (const float*)d_in[5];   // [B,D,D]
    const float* coeffs   = (const float*)d_in[6];   // [R,B]
    const float* attn_vec = (const float*)d_in[7];
    const float* bias     = (const float*)d_in[8];
    const float* fn_g     = (const float*)d_in[9];
    const float* fn_b     = (const float*)d_in[10];
    const float* vals     = (const float*)d_in[11];  // [R,E]
    const int*   rows     = (const int*)d_in[12];    // [R,E]
    const int*   cols     = (const int*)d_in[13];    // [R,E]

    float* ws      = (float*)d_ws;
    float* w_all   = ws;                              // R*D*D
    float* x       = w_all + RR * DD * DD;            // N*D
    float* ping    = x + (size_t)NN * DD;             // N*D
    float* stacked = ping + (size_t)NN * DD;          // R*N*D

    float* outFinal = (float*)d_out;                  // N*D
    float* outAttn  = outFinal + (size_t)NN * DD;     // N*R
    float* outX     = outAttn + (size_t)NN * RR;      // N*D (= node_emb)

    // x_all passthrough
    hipMemcpyAsync(outX, node_emb, (size_t)NN * DD * sizeof(float),
                   hipMemcpyDeviceToDevice, stream);

    // 1) compose relation weights
    compose_kernel<<<(RR * DD * DD) / 256, 256, 0, stream>>>(coeffs, bases, w_all);

    // 2) pre-encoder GEMM (+bias) then LayerNorm -> x
    dim3 gGemm((NN + TILE_M - 1) / TILE_M, 1, 1);
    gemm_xwt_kernel<<<gGemm, 256, 0, stream>>>(node_emb, pre_W, pre_b, ping,
                                               NN, 1, 0, 0);
    layernorm_kernel<<<(NN + 7) / 8, 256, 0, stream>>>(ping, ln1_g, ln1_b, x, NN);

    // 3) 8 relation GEMMs batched over grid.z -> stacked[r] = x @ w_all[r]^T
    dim3 gRel((NN + TILE_M - 1) / TILE_M, 1, RR);
    gemm_xwt_kernel<<<gRel, 256, 0, stream>>>(x, w_all, pre_b, stacked,
                                              NN, 0, DD * DD, (long)NN * DD);

    // 4) per relation: 2 hops of SpMM (scatter-add), ping-pong with stacked[r]
    const size_t hbytes = (size_t)NN * DD * sizeof(float);
    int spmmBlocks = (EE + 31) / 32;  // 8 waves * 4 edges per block
    for (int r = 0; r < RR; ++r) {
        float* hr = stacked + (size_t)r * NN * DD;
        const int*   er = rows + (size_t)r * EE;
        const int*   ec = cols + (size_t)r * EE;
        const float* evp = vals + (size_t)r * EE;
        hipMemsetAsync(ping, 0, hbytes, stream);
        spmm_kernel<<<spmmBlocks, 256, 0, stream>>>(hr, ping, er, ec, evp, EE);
        hipMemsetAsync(hr, 0, hbytes, stream);
        spmm_kernel<<<spmmBlocks, 256, 0, stream>>>(ping, hr, er, ec, evp, EE);
    }

    // 5) fused attention + softmax + GELU + final LayerNorm
    attn_out_kernel<<<(NN + 7) / 8, 256, 0, stream>>>(
        stacked, attn_vec, bias, fn_g, fn_b, outFinal, outAttn, NN);
}